// SelectiveSSM_63290638074648
// MI455X (gfx1250) — compile-verified
//
#include <hip/hip_runtime.h>
#include <hip/hip_bf16.h>
#include <math.h>

// ---------------------------------------------------------------------------
// Selective SSM (Mamba-style) forward for MI455X / gfx1250, wave32, WMMA.
//   B=1, L=2048, D_MODEL=512, D_STATE=16, D_CONV=4, D_INNER=1024
// Pipeline:
//   1) gemm_wmma_f32 : xz = x @ W_in            (2048x512)x(512x2048)  [WMMA f32]
//   2) conv_silu     : depthwise causal conv4 + bias + SiLU
//   3) xproj         : xp = xs @ W_xproj        (N=33, wave-dot kernel)
//   4) scan (3 passes): chunked linear-recurrence scan, 16 chunks of 128
//   5) gemm_wmma_f32 : out = y @ W_out          (2048x1024)x(1024x512) [WMMA f32]
// ---------------------------------------------------------------------------

#define LSEQ    2048
#define DMODEL  512
#define DSTATE  16
#define DINNER  1024
#define NCHUNK  16
#define CHUNK   (LSEQ / NCHUNK)   // 128

typedef float v2f __attribute__((ext_vector_type(2)));
typedef float v8f __attribute__((ext_vector_type(8)));

// Only check in the device pass: host preprocessor cannot see amdgcn builtins.
#if defined(__HIP_DEVICE_COMPILE__)
#if !__has_builtin(__builtin_amdgcn_wmma_f32_16x16x4_f32)
#error "DEVICE: builtin __builtin_amdgcn_wmma_f32_16x16x4_f32 not available"
#endif
#endif

__device__ __forceinline__ float softplus_f(float t) {
    return (t > 20.0f) ? t : log1pf(__expf(t));
}

// ---------------------------------------------------------------------------
// Generic row-major f32 GEMM: C[M,N] = A[M,K] * B[K,N]
// Block tile 64x64, 256 threads (8 waves). Each wave owns a 32x16 strip
// (two 16x16 WMMA accumulators). K staged through LDS in slices of 16.
// Requires M%64==0, N%64==0, K%16==0 (true for all call sites here).
//
// Fragment layouts per CDNA5 ISA 7.12.2 (wave32):
//   A 16x4 : lanes 0-15 -> M=lane, VGPR{0,1}=K{k0,k0+1}; lanes 16-31 -> K{k0+2,k0+3}
//   B 4x16 : VGPR0: lanes0-15 = row k0, lanes16-31 = row k0+2; VGPR1: k0+1 / k0+3
//   C 16x16: VGPR r: lanes0-15 -> (M=r, N=lane), lanes16-31 -> (M=r+8, N=lane-16)
// ---------------------------------------------------------------------------
__global__ __launch_bounds__(256) void gemm_wmma_f32(
    const float* __restrict__ A, const float* __restrict__ B,
    float* __restrict__ C, int M, int N, int K)
{
    __shared__ __align__(16) float As[64][18];   // [m][k], padded (even stride)
    __shared__ __align__(16) float Bs[16][68];   // [k][n], 272B row stride (16B mult)

    const int tid  = threadIdx.x;
    const int lane = tid & 31;
    const int wave = tid >> 5;

    const int m0 = blockIdx.y * 64;
    const int n0 = blockIdx.x * 64;
    const int n0w = (wave & 3) * 16;   // wave's N sub-tile
    const int m0w = (wave >> 2) * 32;  // wave's M strip (two 16-row tiles)

    const int l15   = lane & 15;
    const int khalf = (lane < 16) ? 0 : 2;

    v8f acc0 = {};
    v8f acc1 = {};

    // cooperative tile-load assignments
    const int ar = tid >> 2;          // 0..63  (A row within tile)
    const int aq = (tid & 3) << 2;    // 0,4,8,12 (A k-quad)
    const int br = tid >> 4;          // 0..15  (B k-row)
    const int bq = (tid & 15) << 2;   // 0..60  (B n-quad)

    for (int kb = 0; kb < K; kb += 16) {
        const float4 av = *(const float4*)(A + (size_t)(m0 + ar) * K + kb + aq);
        const float4 bv = *(const float4*)(B + (size_t)(kb + br) * N + n0 + bq);

        if (kb + 16 < K) {  // speculative prefetch of next K slice -> global_prefetch_b8
            __builtin_prefetch(A + (size_t)(m0 + ar) * K + kb + 16 + aq, 0, 1);
            __builtin_prefetch(B + (size_t)(kb + 16 + br) * N + n0 + bq, 0, 1);
        }

        __syncthreads();  // previous slice fully consumed
        As[ar][aq + 0] = av.x;
        As[ar][aq + 1] = av.y;
        As[ar][aq + 2] = av.z;
        As[ar][aq + 3] = av.w;
        *(float4*)&Bs[br][bq] = bv;
        __syncthreads();  // slice visible to all waves

        #pragma unroll
        for (int k0 = 0; k0 < 16; k0 += 4) {
            const int kk = k0 + khalf;
            v2f bf, a0, a1;
            bf.x = Bs[kk    ][n0w + l15];
            bf.y = Bs[kk + 1][n0w + l15];
            a0.x = As[m0w + l15][kk];
            a0.y = As[m0w + l15][kk + 1];
            a1.x = As[m0w + 16 + l15][kk];
            a1.y = As[m0w + 16 + l15][kk + 1];
            acc0 = __builtin_amdgcn_wmma_f32_16x16x4_f32(
                false, a0, false, bf, (short)0, acc0, false, false);
            acc1 = __builtin_amdgcn_wmma_f32_16x16x4_f32(
                false, a1, false, bf, (short)0, acc1, false, false);
        }
    }

    const int crow = m0 + m0w + ((lane < 16) ? 0 : 8);
    const int ccol = n0 + n0w + l15;
    #pragma unroll
    for (int r = 0; r < 8; ++r) {
        C[(size_t)(crow + r) * N + ccol]      = acc0[r];
        C[(size_t)(crow + 16 + r) * N + ccol] = acc1[r];
    }
}

// ---------------------------------------------------------------------------
// Depthwise causal conv (k=4) + bias + SiLU.
// ---------------------------------------------------------------------------
__global__ __launch_bounds__(256) void conv_silu_kernel(
    const float* __restrict__ xz, const float* __restrict__ conv_w,
    const float* __restrict__ conv_b, float* __restrict__ xs_c)
{
    const int idx = blockIdx.x * 256 + threadIdx.x;   // over L*DINNER, d fastest
    const int l = idx / DINNER;
    const int d = idx % DINNER;
    if (l >= LSEQ) return;

    const float w0 = conv_w[d * 4 + 0];
    const float w1 = conv_w[d * 4 + 1];
    const float w2 = conv_w[d * 4 + 2];
    const float w3 = conv_w[d * 4 + 3];

    float acc = conv_b[d];
    acc += w3 * xz[(size_t)l * (2 * DINNER) + d];
    if (l >= 1) acc += w2 * xz[(size_t)(l - 1) * (2 * DINNER) + d];
    if (l >= 2) acc += w1 * xz[(size_t)(l - 2) * (2 * DINNER) + d];
    if (l >= 3) acc += w0 * xz[(size_t)(l - 3) * (2 * DINNER) + d];

    xs_c[idx] = acc / (1.0f + __expf(-acc));   // silu
}

// ---------------------------------------------------------------------------
// xp[l, c] = dot(xs_c[l, :], W_xproj[:, c]),  c in [0, 33)
// ---------------------------------------------------------------------------
__global__ __launch_bounds__(256) void xproj_kernel(
    const float* __restrict__ xs_c, const float* __restrict__ W,
    float* __restrict__ xp)
{
    __shared__ float xrow[DINNER];
    const int l = blockIdx.x;
    for (int i = threadIdx.x; i < DINNER; i += 256)
        xrow[i] = xs_c[(size_t)l * DINNER + i];
    __syncthreads();

    const int wave = threadIdx.x >> 5;
    const int lane = threadIdx.x & 31;
    const int NCOL = 2 * DSTATE + 1;   // 33

    for (int c = wave; c < NCOL; c += 8) {
        float s = 0.0f;
        for (int k = lane; k < DINNER; k += 32)
            s += xrow[k] * W[(size_t)k * NCOL + c];
        s += __shfl_xor(s, 16);
        s += __shfl_xor(s, 8);
        s += __shfl_xor(s, 4);
        s += __shfl_xor(s, 2);
        s += __shfl_xor(s, 1);
        if (lane == 0) xp[(size_t)l * NCOL + c] = s;
    }
}

// ---------------------------------------------------------------------------
// Chunked scan. Recurrence (matching the reference's cumsum formulation, where
// the decay lags one step):  h_l = a_l * h_{l-1} + u_l,
//   a_l = exp(delta_{l-1} * A)  (a_0 = 1),   u_l = delta_l * B_l[s] * x_l.
// a_l depends only on l -> every chunk is self-contained given xp.
//
// Lane mapping (wave32): lanes 0-15 -> channel d0 states 0-15, lanes 16-31 ->
// channel d1. Grid: (64 blocks of 8 waves) x NCHUNK chunks.
// ---------------------------------------------------------------------------

// Pass A: per-chunk transfer pair  P_c = prod(a),  S_c = scan from h=0.
__global__ __launch_bounds__(256) void scan_chunk_state_kernel(
    const float* __restrict__ xs_c, const float* __restrict__ xp,
    const float* __restrict__ dt_w, const float* __restrict__ dt_b,
    const float* __restrict__ A_log, float* __restrict__ Pc,
    float* __restrict__ Sc)
{
    const int lane = threadIdx.x & 31;
    const int wave = threadIdx.x >> 5;
    const int c    = blockIdx.y;
    const int w    = blockIdx.x * 8 + wave;    // 0..511
    const int s    = lane & 15;
    const int d    = w * 2 + (lane >> 4);
    const int NCOL = 2 * DSTATE + 1;
    const int l0   = c * CHUNK;

    const float Aneg = -__expf(A_log[s]);
    const float dtw  = dt_w[d];
    const float dtb  = dt_b[d];

    float delta_prev = 0.0f;
    if (l0 > 0)
        delta_prev = softplus_f(xp[(size_t)(l0 - 1) * NCOL] * dtw + dtb);

    float h = 0.0f, P = 1.0f;
    for (int i = 0; i < CHUNK; ++i) {
        const int l = l0 + i;
        const float* row = xp + (size_t)l * NCOL;
        const float dtr  = row[0];
        const float b    = row[1 + s];
        const float x    = xs_c[(size_t)l * DINNER + d];
        const float delta = softplus_f(dtr * dtw + dtb);
        const float a     = (l == 0) ? 1.0f : __expf(delta_prev * Aneg);
        h = a * h + delta * b * x;
        P *= a;
        delta_prev = delta;
    }
    const size_t idx = ((size_t)c * DINNER + d) * DSTATE + s;
    Pc[idx] = P;
    Sc[idx] = h;
}

// Pass B: sequential combine over the 16 chunk summaries -> chunk entry states.
__global__ __launch_bounds__(256) void scan_combine_kernel(
    const float* __restrict__ Pc, const float* __restrict__ Sc,
    float* __restrict__ hin)
{
    const int tid = blockIdx.x * 256 + threadIdx.x;   // (d,s) pair, 0..16383
    if (tid >= DINNER * DSTATE) return;
    float h = 0.0f;
    for (int c = 0; c < NCHUNK; ++c) {
        const size_t idx = (size_t)c * DINNER * DSTATE + tid;
        hin[idx] = h;                  // state entering chunk c
        h = Pc[idx] * h + Sc[idx];
    }
}

// Pass C: re-run each chunk from hin, emit y = (C.h + D*x) * silu(z).
__global__ __launch_bounds__(256) void scan_emit_kernel(
    const float* __restrict__ xz, const float* __restrict__ xs_c,
    const float* __restrict__ xp, const float* __restrict__ dt_w,
    const float* __restrict__ dt_b, const float* __restrict__ A_log,
    const float* __restrict__ Dvec, const float* __restrict__ hin,
    float* __restrict__ y_g)
{
    const int lane = threadIdx.x & 31;
    const int wave = threadIdx.x >> 5;
    const int c    = blockIdx.y;
    const int w    = blockIdx.x * 8 + wave;    // 0..511
    const int s    = lane & 15;
    const int d    = w * 2 + (lane >> 4);
    const int NCOL = 2 * DSTATE + 1;
    const int l0   = c * CHUNK;

    const float Aneg = -__expf(A_log[s]);
    const float dtw  = dt_w[d];
    const float dtb  = dt_b[d];
    const float Dd   = Dvec[d];

    float delta_prev = 0.0f;
    if (l0 > 0)
        delta_prev = softplus_f(xp[(size_t)(l0 - 1) * NCOL] * dtw + dtb);

    float h = hin[((size_t)c * DINNER + d) * DSTATE + s];

    for (int i = 0; i < CHUNK; ++i) {
        const int l = l0 + i;
        const float* row = xp + (size_t)l * NCOL;
        const float dtr  = row[0];
        const float b    = row[1 + s];
        const float cc   = row[1 + DSTATE + s];
        const float x    = xs_c[(size_t)l * DINNER + d];
        const float z    = xz[(size_t)l * (2 * DINNER) + DINNER + d];

        const float delta = softplus_f(dtr * dtw + dtb);
        const float a     = (l == 0) ? 1.0f : __expf(delta_prev * Aneg);
        h = a * h + delta * b * x;
        delta_prev = delta;

        float p = cc * h;
        p += __shfl_xor(p, 1);
        p += __shfl_xor(p, 2);
        p += __shfl_xor(p, 4);
        p += __shfl_xor(p, 8);

        if (s == 0) {
            const float y  = p + Dd * x;
            const float sz = z / (1.0f + __expf(-z));
            y_g[(size_t)l * DINNER + d] = y * sz;
        }
    }
}

// ---------------------------------------------------------------------------
// Launch
// ---------------------------------------------------------------------------
extern "C" void kernel_launch(void* const* d_in, const int* in_sizes, int n_in,
                              void* d_out, int out_size, void* d_ws, size_t ws_size,
                              hipStream_t stream)
{
    const float* x      = (const float*)d_in[0];
    const float* W_in   = (const float*)d_in[1];
    const float* conv_w = (const float*)d_in[2];
    const float* conv_b = (const float*)d_in[3];
    const float* W_xprj = (const float*)d_in[4];
    const float* dt_w   = (const float*)d_in[5];
    const float* dt_b   = (const float*)d_in[6];
    const float* A_log  = (const float*)d_in[7];
    const float* Dvec   = (const float*)d_in[8];
    const float* W_out  = (const float*)d_in[9];
    float* out = (float*)d_out;

    // workspace layout (floats)
    float* ws   = (float*)d_ws;
    float* xz   = ws;                                    // 2048*2048
    float* xs_c = xz + (size_t)LSEQ * 2 * DINNER;        // 2048*1024
    float* xp   = xs_c + (size_t)LSEQ * DINNER;          // 2048*33
    float* y_g  = xp + (size_t)LSEQ * (2 * DSTATE + 1);  // 2048*1024
    float* Pc   = y_g + (size_t)LSEQ * DINNER;           // 16*1024*16
    float* Sc   = Pc + (size_t)NCHUNK * DINNER * DSTATE; // 16*1024*16
    float* hin  = Sc + (size_t)NCHUNK * DINNER * DSTATE; // 16*1024*16

    // 1) xz = x @ W_in : M=2048, N=2048, K=512
    {
        dim3 grid((2 * DINNER) / 64, LSEQ / 64);
        gemm_wmma_f32<<<grid, 256, 0, stream>>>(x, W_in, xz, LSEQ, 2 * DINNER, DMODEL);
    }
    // 2) depthwise conv + SiLU
    conv_silu_kernel<<<(LSEQ * DINNER) / 256, 256, 0, stream>>>(xz, conv_w, conv_b, xs_c);
    // 3) xp = xs_c @ W_xproj
    xproj_kernel<<<LSEQ, 256, 0, stream>>>(xs_c, W_xprj, xp);
    // 4) chunked SSM scan
    {
        dim3 grid(64, NCHUNK);   // 64 blocks x 8 waves = 512 waves per chunk
        scan_chunk_state_kernel<<<grid, 256, 0, stream>>>(
            xs_c, xp, dt_w, dt_b, A_log, Pc, Sc);
        scan_combine_kernel<<<(DINNER * DSTATE) / 256, 256, 0, stream>>>(Pc, Sc, hin);
        scan_emit_kernel<<<grid, 256, 0, stream>>>(
            xz, xs_c, xp, dt_w, dt_b, A_log, Dvec, hin, y_g);
    }
    // 5) out = y_g @ W_out : M=2048, N=512, K=1024
    {
        dim3 grid(DMODEL / 64, LSEQ / 64);
        gemm_wmma_f32<<<grid, 256, 0, stream>>>(y_g, W_out, out, LSEQ, DMODEL, DINNER);
    }
    (void)in_sizes; (void)n_in; (void)out_size; (void)ws_size;
}